// CORAL_16080357556585
// MI455X (gfx1250) — compile-verified
//
#include <hip/hip_runtime.h>
#include <hip/hip_bf16.h>

// CORAL loss: mean -> centered covariance (fp32 WMMA SYRK, 8-accumulator
// register blocking, K-pair-interleaved LDS so every fragment is a single
// ds_load_b64) -> cov Gram (fp32 WMMA) -> masked pairwise reduction.

typedef __attribute__((ext_vector_type(2))) float v2f;
typedef __attribute__((ext_vector_type(8))) float v8f;

#define BZ   32
#define NN   4096
#define DD   256
// K2: K-pair interleaved tile.  16 pair-rows (32 K) x 128 dims x 2 dwords,
// pair-row stride 288 dwords (288 % 64 == 32 -> lane halves use disjoint banks).
#define PS   288
// K3: row stride 132 dwords (132 % 64 == 4 -> conflict-free b64 column reads).
#define SB   132

__device__ __forceinline__ v2f lds_v2f(const float* p) {
    return *(const v2f*)p;   // 8B-aligned by construction -> ds_load_b64
}

// ---------------------------------------------------------------- K1: means
__global__ void coral_mean(const float* __restrict__ feat, float* __restrict__ mean) {
    const int b = blockIdx.x;          // 32 blocks
    const int t = threadIdx.x;         // 256 threads, one per dim
    const float* fb = feat + (size_t)b * NN * DD;
    float s = 0.0f;
    for (int n = 0; n < NN; ++n) s += fb[(size_t)n * DD + t];
    mean[b * DD + t] = s * (1.0f / (float)NN);
}

// ------------------------------------------- K2: cov[b] = cent^T cent / (N-1)
// grid: 32 batches * (2x2) tiles of 128x128.  256 threads = 8 waves.
// Wave w owns a 32x64 slab: 2 row subtiles x 4 col subtiles = 8 accumulators.
// Per k-step: 6 single ds_load_b64 fragments feed 8 WMMAs.
__global__ void __launch_bounds__(256)
coral_cov(const float* __restrict__ feat,
          const float* __restrict__ mean,
          float* __restrict__ cov) {
    __shared__ __align__(16) float As[16 * PS];
    __shared__ __align__(16) float Bs[16 * PS];

    const int b    = blockIdx.x >> 2;
    const int ti   = (blockIdx.x >> 1) & 1;   // output row tile (dims I, 128-wide)
    const int tj   = blockIdx.x & 1;          // output col tile (dims J, 128-wide)
    const int t    = threadIdx.x;
    const int wave = t >> 5;
    const int lane = t & 31;
    const int c    = t & 127;                 // loader column (0..127)
    const int r2   = t >> 7;                  // loader row parity (0..1)

    const float mA = mean[b * DD + ti * 128 + c];
    const float mB = mean[b * DD + tj * 128 + c];
    const float* fb = feat + (size_t)b * NN * DD;

    // wave -> 32-row strip x 64-col half of the 128x128 tile
    const int rbase = (wave & 3) * 32;        // 0,32,64,96
    const int cbase = (wave >> 2) * 64;       // 0 or 64
    const int m     = lane & 15;              // fragment row/col within 16
    const int khalf = (lane >> 4) * 2;        // K offset per lane half (A/B layout)

    v8f acc[2][4];
#pragma unroll
    for (int i = 0; i < 2; ++i)
#pragma unroll
        for (int j = 0; j < 4; ++j) acc[i][j] = (v8f){};

    // staging address: element (k, d) -> (k>>1)*PS + 2*d + (k&1)
    const int stA = r2;                       // (row & 1) == r2 by construction
    for (int k0 = 0; k0 < NN; k0 += 32) {
#pragma unroll
        for (int rr = 0; rr < 16; ++rr) {
            const int row = rr * 2 + r2;      // row>>1 == rr
            const size_t gbase = (size_t)(k0 + row) * DD;
            As[rr * PS + c * 2 + stA] = fb[gbase + ti * 128 + c] - mA;
            Bs[rr * PS + c * 2 + stA] = fb[gbase + tj * 128 + c] - mB;
        }
        __syncthreads();

#pragma unroll
        for (int kk = 0; kk < 32; kk += 4) {
            const int kp = (kk + khalf) >> 1;  // pair-row index (kb is even)
            v2f a0, a1, bf[4];
            // A fragments: rows kb,kb+1 at column (rbase(+16)+m) -> one b64
            a0 = lds_v2f(&As[kp * PS + (rbase + m) * 2]);
            a1 = lds_v2f(&As[kp * PS + (rbase + 16 + m) * 2]);
#pragma unroll
            for (int j = 0; j < 4; ++j)
                bf[j] = lds_v2f(&Bs[kp * PS + (cbase + j * 16 + m) * 2]);
#pragma unroll
            for (int j = 0; j < 4; ++j) {
                acc[0][j] = __builtin_amdgcn_wmma_f32_16x16x4_f32(
                    false, a0, false, bf[j], (short)0, acc[0][j], false, false);
                acc[1][j] = __builtin_amdgcn_wmma_f32_16x16x4_f32(
                    false, a1, false, bf[j], (short)0, acc[1][j], false, false);
            }
        }
        __syncthreads();
    }

    // C/D layout: VGPR r -> M=r (lanes 0-15), M=r+8 (lanes 16-31); N = lane&15
    const float scale = 1.0f / (float)(NN - 1);
    const int rowBase = ti * 128 + rbase + ((lane >> 4) ? 8 : 0);
    const int colBase = tj * 128 + cbase + m;
    float* cb_out = cov + (size_t)b * DD * DD;
#pragma unroll
    for (int i = 0; i < 2; ++i)
#pragma unroll
        for (int j = 0; j < 4; ++j)
#pragma unroll
            for (int r = 0; r < 8; ++r) {
                const int row = rowBase + i * 16 + r;
                const int col = colBase + j * 16;
                cb_out[(size_t)row * DD + col] = acc[i][j][r] * scale;
            }
}

// -------------------------------- K3: G = V V^T, V = cov viewed as 32x65536
// 32 blocks, each owns a disjoint K range of 2048 and writes a partial 32x32.
__global__ void coral_gram(const float* __restrict__ cov, float* __restrict__ gp) {
    __shared__ __align__(16) float Cs[32 * SB];

    const int p    = blockIdx.x;            // 32 partials
    const int t    = threadIdx.x;           // 256 threads = 8 waves
    const int wave = t >> 5;
    const int lane = t & 31;
    const int m    = lane & 15;
    const int khalf = (lane >> 4) * 2;

    const int sr = (wave >> 1) & 1;         // waves 0..3 -> 2x2 subtiles
    const int sc = wave & 1;

    v8f acc = {};

    for (int q = 0; q < 16; ++q) {
        const int kg = p * 2048 + q * 128;
        // stage 32 rows x 128 K into LDS (coalesced)
#pragma unroll
        for (int e = 0; e < 16; ++e) {
            const int idx = e * 256 + t;
            const int row = idx >> 7;
            const int col = idx & 127;
            Cs[row * SB + col] = cov[(size_t)row * (DD * DD) + kg + col];
        }
        __syncthreads();

        if (wave < 4) {                     // wave-uniform: EXEC stays all-ones
#pragma unroll
            for (int kk = 0; kk < 128; kk += 4) {
                const int kb = kk + khalf;  // even -> 8B aligned
                v2f a  = lds_v2f(&Cs[(sr * 16 + m) * SB + kb]);
                v2f bb = lds_v2f(&Cs[(sc * 16 + m) * SB + kb]);
                acc = __builtin_amdgcn_wmma_f32_16x16x4_f32(false, a, false, bb,
                                                            (short)0, acc, false, false);
            }
        }
        __syncthreads();
    }

    if (wave < 4) {
        const int rowBase = sr * 16 + ((lane >> 4) ? 8 : 0);
        const int col     = sc * 16 + m;
#pragma unroll
        for (int r = 0; r < 8; ++r)
            gp[p * 1024 + (rowBase + r) * 32 + col] = acc[r];
    }
}

// ------------------------- K4: combine partials, masked pairwise loss -> out
__global__ void coral_final(const float* __restrict__ mean,
                            const float* __restrict__ gp,
                            const long long* __restrict__ dom,
                            float* __restrict__ out) {
    __shared__ float sG[1024];
    __shared__ float sval[1024];
    __shared__ int   scnt[1024];

    const int t = threadIdx.x;              // 1024 threads, one per (i,j)
    float g = 0.0f;
    for (int p = 0; p < 32; ++p) g += gp[p * 1024 + t];
    sG[t] = g;
    __syncthreads();

    const int i = t >> 5;
    const int j = t & 31;
    float val = 0.0f;
    int   cnt = 0;
    if (i < j && dom[i] != dom[j]) {
        float md = 0.0f;
        for (int d = 0; d < DD; ++d) {
            const float diff = mean[i * DD + d] - mean[j * DD + d];
            md += diff * diff;
        }
        md *= (1.0f / (float)DD);
        const float cd = (sG[i * 32 + i] + sG[j * 32 + j] - 2.0f * sG[i * 32 + j])
                         * (1.0f / (float)(DD * DD));
        val = md + cd;
        cnt = 1;
    }
    sval[t] = val;
    scnt[t] = cnt;
    __syncthreads();

    for (int s = 512; s > 0; s >>= 1) {
        if (t < s) { sval[t] += sval[t + s]; scnt[t] += scnt[t + s]; }
        __syncthreads();
    }
    if (t == 0) {
        float loss = sval[0];
        const int num = scnt[0];
        if (num > 1) loss /= (float)num;
        out[0] = loss;
    }
}

extern "C" void kernel_launch(void* const* d_in, const int* in_sizes, int n_in,
                              void* d_out, int out_size, void* d_ws, size_t ws_size,
                              hipStream_t stream) {
    (void)in_sizes; (void)n_in; (void)out_size; (void)ws_size;
    const float*     feat = (const float*)d_in[0];
    const long long* dom  = (const long long*)d_in[1];   // int64 domains
    float* out = (float*)d_out;

    // workspace layout (floats): mean[32*256] | cov[32*256*256] | gp[32*1024]
    float* ws   = (float*)d_ws;
    float* mean = ws;
    float* cov  = ws + (size_t)BZ * DD;
    float* gp   = cov + (size_t)BZ * DD * DD;

    coral_mean <<<BZ,       256, 0, stream>>>(feat, mean);
    coral_cov  <<<BZ * 4,   256, 0, stream>>>(feat, mean, cov);
    coral_gram <<<32,       256, 0, stream>>>(cov, gp);
    coral_final<<<1,       1024, 0, stream>>>(mean, gp, dom, out);
}